// diffusionReaction_72206990180574
// MI455X (gfx1250) — compile-verified
//
#include <hip/hip_runtime.h>
#include <cmath>

// ---------------------------------------------------------------------------
// Diffusion-reaction on 512x512x4, 8 steps:
//   step: per-channel CG solve of (L + g I) x = g u  (10 fixed iters),
//         then U += DT * tanh(tanh(tanh(U K1) K2) K3)
//
// MI455X reasoning: total work ~3-4 GFLOP, working set ~21 MB (L2-resident,
// 192 MB L2) -> the bottleneck is the serial CG dependency chain, not HBM or
// compute. So: minimal number of serially-dependent launches (p-update fused
// into the stencil kernel via ping-pong p buffers: 4 launches/CG iter),
// deterministic two-pass reductions (no fp atomics -> graph-safe), and the
// MLP mapped onto V_WMMA_F32_16X16X4_F32 (16x4 @ 4x16 is the native CDNA5
// f32 WMMA shape; 9 WMMAs per tile of 16 rows).
// ---------------------------------------------------------------------------

#define NXg 512
#define NYg 512
#define NCh 4
#define NPTS (NXg * NYg)
static constexpr float H2INV = 256.0f;      // 1/h^2, h = 1/16
static constexpr float DTc   = 1.0f / 64.0f;

typedef __attribute__((ext_vector_type(2))) float v2f;
typedef __attribute__((ext_vector_type(8))) float v8f;

// block = 256 threads
__device__ __forceinline__ float blockReduce(float v, float* sm) {
  const int tid = threadIdx.x;
  sm[tid] = v;
  __syncthreads();
#pragma unroll
  for (int s = 128; s > 0; s >>= 1) {
    if (tid < s) sm[tid] += sm[tid + s];
    __syncthreads();
  }
  return sm[0];
}

// b = gamma*U ; x = 0 ; r = b ; pA = b (read as p_old with beta=0 in iter 0)
__global__ __launch_bounds__(256) void kInit(const float* __restrict__ Uin,
                                             const float* __restrict__ scale,
                                             float* __restrict__ x,
                                             float* __restrict__ r,
                                             float* __restrict__ pA,
                                             float* __restrict__ partials) {
  __shared__ float sm[256];
  const int c = blockIdx.y;
  const int n = blockIdx.x * 256 + threadIdx.x;
  const float gamma = 1.0f / (scale[c] * DTc);
  const float b = gamma * Uin[n * NCh + c];
  const int idx = c * NPTS + n;
  x[idx]  = 0.0f;
  r[idx]  = b;
  pA[idx] = b;
  const float s = blockReduce(b * b, sm);
  if (threadIdx.x == 0) partials[c * gridDim.x + blockIdx.x] = s;
}

// Fused: p_new = r + beta*p_old (recomputed at all 5 stencil points),
//        Ap = lap(p_new)/h^2 + gamma*p_new, partial dot(p_new, Ap).
// p_old is read-only and p_new is a distinct buffer -> no cross-block race.
__global__ __launch_bounds__(256) void kApplyDotP(const float* __restrict__ r,
                                                  const float* __restrict__ pold,
                                                  const float* __restrict__ scale,
                                                  const float* __restrict__ scalars,
                                                  float* __restrict__ pnew,
                                                  float* __restrict__ Ap,
                                                  float* __restrict__ partials) {
  __shared__ float sm[256];
  const int c = blockIdx.y;
  const int n = blockIdx.x * 256 + threadIdx.x;
  const int gx = n & (NXg - 1);
  const int gy = n >> 9;
  const float beta = scalars[8 + c];
  const float* rc = r    + c * NPTS;
  const float* pc = pold + c * NPTS;
  const float ctr = rc[n] + beta * pc[n];
  float lap = 0.0f;
  if (gx > 0)       lap += ctr - (rc[n - 1]   + beta * pc[n - 1]);
  if (gx < NXg - 1) lap += ctr - (rc[n + 1]   + beta * pc[n + 1]);
  if (gy > 0)       lap += ctr - (rc[n - NXg] + beta * pc[n - NXg]);
  if (gy < NYg - 1) lap += ctr - (rc[n + NXg] + beta * pc[n + NXg]);
  const float gamma = 1.0f / (scale[c] * DTc);
  const float ap = lap * H2INV + gamma * ctr;
  pnew[c * NPTS + n] = ctr;
  Ap[c * NPTS + n]   = ap;
  const float s = blockReduce(ctr * ap, sm);
  if (threadIdx.x == 0) partials[c * gridDim.x + blockIdx.x] = s;
}

// mode 0: rs = sum, beta = 0      (seed so iter 0 gives p = r = b)
// mode 1: alpha = rs/sum
// mode 2: beta = sum/rs ; rs = sum
// scalars: [0..3]=rs  [4..7]=alpha  [8..11]=beta
__global__ __launch_bounds__(256) void kReduce(const float* __restrict__ partials,
                                               int nblk, float* __restrict__ scalars,
                                               int mode) {
  __shared__ float sm[256];
  for (int c = 0; c < NCh; ++c) {
    float s = 0.0f;
    for (int i = threadIdx.x; i < nblk; i += 256) s += partials[c * nblk + i];
    s = blockReduce(s, sm);
    if (threadIdx.x == 0) {
      if (mode == 0)    { scalars[c] = s; scalars[8 + c] = 0.0f; }
      else if (mode == 1) scalars[4 + c] = scalars[c] / s;
      else              { scalars[8 + c] = s / scalars[c]; scalars[c] = s; }
    }
    __syncthreads();
  }
}

// x += alpha p ; r -= alpha Ap ; partial dot(r,r)
__global__ __launch_bounds__(256) void kUpdateXR(float* __restrict__ x,
                                                 float* __restrict__ r,
                                                 const float* __restrict__ pv,
                                                 const float* __restrict__ Ap,
                                                 const float* __restrict__ scalars,
                                                 float* __restrict__ partials) {
  __shared__ float sm[256];
  const int c = blockIdx.y;
  const int n = blockIdx.x * 256 + threadIdx.x;
  const int idx = c * NPTS + n;
  const float alpha = scalars[4 + c];
  x[idx] += alpha * pv[idx];
  const float rn = r[idx] - alpha * Ap[idx];
  r[idx] = rn;
  const float s = blockReduce(rn * rn, sm);
  if (threadIdx.x == 0) partials[c * gridDim.x + blockIdx.x] = s;
}

// MLP: one wave32 per 16-row tile. xsol is channel-major (c*NPTS + row).
// Layer1: (16x4)@(4x16) = one V_WMMA_F32_16X16X4_F32.
// Layers 2/3: 4 chained K=4 WMMAs; C-layout -> A-layout via per-wave LDS tile.
// A 16x4 f32 layout (ISA 7.12.2): lanes 0-15 hold K={0,1}, lanes 16-31 K={2,3};
// B symmetric (K pairs split across lane halves, N = lane%16).
__global__ __launch_bounds__(256) void kMLP(const float* __restrict__ xsol,
                                            const float* __restrict__ K1,
                                            const float* __restrict__ K2,
                                            const float* __restrict__ K3,
                                            float* __restrict__ dst) {
  __shared__ float tile[8 * 256];            // 8 waves x (16x16) f32
  const int wave  = threadIdx.x >> 5;
  const int lane  = threadIdx.x & 31;
  const int laneM = lane & 15;
  const int hi    = lane >> 4;               // 0: K low half, 1: K high half
  const int row0  = blockIdx.x * 128 + wave * 16;
  float* t = tile + wave * 256;

  // ---- layer 1 ----
  v2f a, b;
  a.x = xsol[(2 * hi + 0) * NPTS + row0 + laneM];
  a.y = xsol[(2 * hi + 1) * NPTS + row0 + laneM];
  b.x = K1[(2 * hi + 0) * 16 + laneM];
  b.y = K1[(2 * hi + 1) * 16 + laneM];
  v8f acc = {};
  acc = __builtin_amdgcn_wmma_f32_16x16x4_f32(false, a, false, b, (short)0, acc,
                                              false, false);
#pragma unroll
  for (int v = 0; v < 8; ++v)                // C/D layout: (M = v+8*hi, N = laneM)
    t[(v + 8 * hi) * 16 + laneM] = tanhf(acc[v]);
  __syncthreads();

  // ---- layer 2: (16x16)@(16x16) as 4 x K=4 ----
  v8f acc2 = {};
#pragma unroll
  for (int k = 0; k < 4; ++k) {
    v2f a2, b2;
    a2.x = t[laneM * 16 + 4 * k + 2 * hi + 0];
    a2.y = t[laneM * 16 + 4 * k + 2 * hi + 1];
    b2.x = K2[(4 * k + 2 * hi + 0) * 16 + laneM];
    b2.y = K2[(4 * k + 2 * hi + 1) * 16 + laneM];
    acc2 = __builtin_amdgcn_wmma_f32_16x16x4_f32(false, a2, false, b2, (short)0,
                                                 acc2, false, false);
  }
  __syncthreads();                            // WAR on tile
#pragma unroll
  for (int v = 0; v < 8; ++v)
    t[(v + 8 * hi) * 16 + laneM] = tanhf(acc2[v]);
  __syncthreads();

  // ---- layer 3: (16x16)@(16x4 zero-padded to 16x16) ----
  v8f acc3 = {};
#pragma unroll
  for (int k = 0; k < 4; ++k) {
    v2f a3, b3;
    a3.x = t[laneM * 16 + 4 * k + 2 * hi + 0];
    a3.y = t[laneM * 16 + 4 * k + 2 * hi + 1];
    b3.x = (laneM < NCh) ? K3[(4 * k + 2 * hi + 0) * NCh + laneM] : 0.0f;
    b3.y = (laneM < NCh) ? K3[(4 * k + 2 * hi + 1) * NCh + laneM] : 0.0f;
    acc3 = __builtin_amdgcn_wmma_f32_16x16x4_f32(false, a3, false, b3, (short)0,
                                                 acc3, false, false);
  }

  // U_new = U_diff + DT*tanh(.) ; only N(=laneM) < 4 columns are real
  if (laneM < NCh) {
#pragma unroll
    for (int v = 0; v < 8; ++v) {
      const int row = row0 + v + 8 * hi;
      const float u = xsol[laneM * NPTS + row];
      dst[row * NCh + laneM] = u + DTc * tanhf(acc3[v]);
    }
  }
}

extern "C" void kernel_launch(void* const* d_in, const int* in_sizes, int n_in,
                              void* d_out, int out_size, void* d_ws, size_t ws_size,
                              hipStream_t stream) {
  (void)in_sizes; (void)n_in; (void)out_size; (void)ws_size;
  const float* U0    = (const float*)d_in[0];
  const float* scale = (const float*)d_in[1];
  const float* K1    = (const float*)d_in[2];
  const float* K2    = (const float*)d_in[3];
  const float* K3    = (const float*)d_in[4];
  float* out = (float*)d_out;

  // workspace layout (floats): x, r, pA, pB, Ap, Ubuf each 4*NPTS; partials;
  // scalars (12 used). Total ~25 MB.
  float* ws = (float*)d_ws;
  float* x        = ws;
  float* r        = x  + 4 * NPTS;
  float* pA       = r  + 4 * NPTS;
  float* pB       = pA + 4 * NPTS;
  float* Ap       = pB + 4 * NPTS;
  float* Ub       = Ap + 4 * NPTS;
  float* partials = Ub + 4 * NPTS;                 // NCh * (NPTS/256)
  float* scalars  = partials + NCh * (NPTS / 256); // 12 used

  const dim3 grid2(NPTS / 256, NCh);
  const int  nblk = NPTS / 256;

  for (int step = 0; step < 8; ++step) {
    const float* Uin  = (step == 0) ? U0  : Ub;
    float*       Uout = (step == 7) ? out : Ub;

    kInit<<<grid2, 256, 0, stream>>>(Uin, scale, x, r, pA, partials);
    kReduce<<<1, 256, 0, stream>>>(partials, nblk, scalars, 0);   // rs, beta=0
    for (int it = 0; it < 10; ++it) {
      float* pold = (it & 1) ? pB : pA;
      float* pnew = (it & 1) ? pA : pB;
      kApplyDotP<<<grid2, 256, 0, stream>>>(r, pold, scale, scalars,
                                            pnew, Ap, partials);
      kReduce<<<1, 256, 0, stream>>>(partials, nblk, scalars, 1); // alpha
      kUpdateXR<<<grid2, 256, 0, stream>>>(x, r, pnew, Ap, scalars, partials);
      kReduce<<<1, 256, 0, stream>>>(partials, nblk, scalars, 2); // beta, rs
    }
    kMLP<<<NPTS / 128, 256, 0, stream>>>(x, K1, K2, K3, Uout);
  }
}